// ActorCriticSAGE_78030965834147
// MI455X (gfx1250) — compile-verified
//
#include <hip/hip_runtime.h>

typedef __attribute__((ext_vector_type(2))) float v2f;
typedef __attribute__((ext_vector_type(8))) float v8f;

#define D 128
#define LDA 130   // LDS row stride: even (8B-aligned float2) and bank-conflict-free

// ---------------- utility ----------------
__global__ void zero_f32(float* __restrict__ p, long n) {
    long i = (long)blockIdx.x * blockDim.x + threadIdx.x;
    if (i < n) p[i] = 0.0f;
}

// ---------------- degree ----------------
__global__ void degree_kernel(const int* __restrict__ dst, float* __restrict__ deg, int E) {
    int e = blockIdx.x * blockDim.x + threadIdx.x;
    if (e < E) atomicAdd(&deg[dst[e]], 1.0f);
}

// ---------------- edge scatter: agg[dst] += x[src] ----------------
// one 32-lane group per edge, 4 floats per lane
__global__ void scatter_kernel(const float* __restrict__ x,
                               const int* __restrict__ src,
                               const int* __restrict__ dst,
                               float* __restrict__ agg, int E) {
    long t = (long)blockIdx.x * blockDim.x + threadIdx.x;
    int e = (int)(t >> 5);
    int l = (int)(t & 31);
    if (e < E) {
        int s = src[e], d = dst[e];
        const float4 v = *(const float4*)(x + (long)s * D + l * 4);
        float* a = agg + (long)d * D + l * 4;
        atomicAdd(a + 0, v.x);
        atomicAdd(a + 1, v.y);
        atomicAdd(a + 2, v.z);
        atomicAdd(a + 3, v.w);
    }
}

// ---------------- fused SAGE layer GEMM (fp32 WMMA, exact fp32 math) ----------------
// out[m, :] = relu( (agg[m,:]/max(deg,1)) @ Wl^T + bias + xin[m,:] @ Wr^T )
// block = 256 threads (8 waves) handles a 16-row strip; wave w owns cols [16w,16w+16)
// `out` MAY alias `agg`: all agg reads happen before the barrier; rows are block-private.
__global__ __launch_bounds__(256) void sage_gemm_kernel(
        const float* __restrict__ agg, const float* __restrict__ deg,
        const float* __restrict__ xin,
        const float* __restrict__ Wl, const float* __restrict__ bias,
        const float* __restrict__ Wr, float* __restrict__ out) {
    __shared__ float sMean[16 * LDA];
    __shared__ float sX[16 * LDA];
    __shared__ float sDegInv[16];

    const int m0 = blockIdx.x * 16;
    const int t  = threadIdx.x;

    if (t < 16) sDegInv[t] = 1.0f / fmaxf(deg[m0 + t], 1.0f);
    __syncthreads();

    // stage 16x128 of mean-agg and of xin into LDS (8 elements / thread each)
#pragma unroll
    for (int j = 0; j < 8; ++j) {
        int idx = t + j * 256;
        int r = idx >> 7, c = idx & 127;
        long g = (long)(m0 + r) * D + c;
        sMean[r * LDA + c] = agg[g] * sDegInv[r];
        sX[r * LDA + c]    = xin[g];
    }
    __syncthreads();

    const int lane = t & 31;
    const int wave = t >> 5;
    const int n0   = wave * 16;
    const int hi   = lane >> 4;        // 0: lanes 0-15, 1: lanes 16-31
    const int mrow = lane & 15;        // A-matrix row owned by this lane
    const int n    = n0 + (lane & 15); // B/C column owned by this lane
    const int kb   = hi * 2;           // K sub-offset per ISA 32-bit A/B layout

    const float* wlrow = Wl + (long)n * D;  // B[k][n] = W[n][k]
    const float* wrrow = Wr + (long)n * D;

    v8f c;
    const float bv = bias[n];
#pragma unroll
    for (int r = 0; r < 8; ++r) c[r] = bv;

    // mean_agg @ Wl^T  : K = 128 in steps of 4
#pragma unroll
    for (int kk = 0; kk < 32; ++kk) {
        int k = kk * 4 + kb;
        v2f a = *(const v2f*)(&sMean[mrow * LDA + k]);
        v2f b = *(const v2f*)(wlrow + k);
        c = __builtin_amdgcn_wmma_f32_16x16x4_f32(false, a, false, b,
                                                  (short)0, c, false, false);
    }
    // xin @ Wr^T
#pragma unroll
    for (int kk = 0; kk < 32; ++kk) {
        int k = kk * 4 + kb;
        v2f a = *(const v2f*)(&sX[mrow * LDA + k]);
        v2f b = *(const v2f*)(wrrow + k);
        c = __builtin_amdgcn_wmma_f32_16x16x4_f32(false, a, false, b,
                                                  (short)0, c, false, false);
    }

    // C/D layout: VGPR r -> M = r (+8 for lanes 16-31), N = lane&15
#pragma unroll
    for (int r = 0; r < 8; ++r) {
        int m = m0 + r + hi * 8;
        out[(long)m * D + n] = fmaxf(c[r], 0.0f);
    }
}

// ---------------- actor head: out[i] = h[i,:] . Wa + ba ----------------
__global__ void actor_kernel(const float* __restrict__ h,
                             const float* __restrict__ Wa, const float* __restrict__ ba,
                             float* __restrict__ out, int N) {
    int node = blockIdx.x * 8 + (threadIdx.x >> 5);
    int lane = threadIdx.x & 31;
    if (node >= N) return;
    float4 hv = *(const float4*)(h + (long)node * D + lane * 4);
    float4 wv = *(const float4*)(Wa + lane * 4);
    float p = hv.x * wv.x + hv.y * wv.y + hv.z * wv.z + hv.w * wv.w;
#pragma unroll
    for (int off = 16; off > 0; off >>= 1) p += __shfl_xor(p, off, 32);
    if (lane == 0) out[node] = p + ba[0];
}

// ---------------- critic: column sums then tiny reduction ----------------
__global__ void critic_partial_kernel(const float* __restrict__ h,
                                      float* __restrict__ colsum, int N) {
    int c = threadIdx.x;                 // 0..127
    int start = blockIdx.x * 256;
    int end = min(start + 256, N);
    float s = 0.0f;
    for (int i = start; i < end; ++i) s += h[(long)i * D + c];
    atomicAdd(&colsum[c], s);
}

__global__ void critic_final_kernel(const float* __restrict__ colsum,
                                    const float* __restrict__ Wc,
                                    const float* __restrict__ bc,
                                    float* __restrict__ out, float invN) {
    __shared__ float red[128];
    int c = threadIdx.x;
    red[c] = colsum[c] * Wc[c];
    __syncthreads();
    for (int s = 64; s > 0; s >>= 1) {
        if (c < s) red[c] += red[c + s];
        __syncthreads();
    }
    if (c == 0) out[0] = red[0] * invN + bc[0];
}

// ---------------- host launcher ----------------
extern "C" void kernel_launch(void* const* d_in, const int* in_sizes, int n_in,
                              void* d_out, int out_size, void* d_ws, size_t ws_size,
                              hipStream_t stream) {
    const float* x   = (const float*)d_in[0];
    const int*   ei  = (const int*)d_in[1];
    const float* W1l = (const float*)d_in[2];
    const float* b1  = (const float*)d_in[3];
    const float* W1r = (const float*)d_in[4];
    const float* W2l = (const float*)d_in[5];
    const float* b2  = (const float*)d_in[6];
    const float* W2r = (const float*)d_in[7];
    const float* Wa  = (const float*)d_in[8];
    const float* ba  = (const float*)d_in[9];
    const float* Wc  = (const float*)d_in[10];
    const float* bc  = (const float*)d_in[11];

    const int N = in_sizes[0] / D;   // 100000 (divisible by 16)
    const int E = in_sizes[1] / 2;   // 1600000
    const int* src = ei;
    const int* dst = ei + E;

    float* ws     = (float*)d_ws;
    float* deg    = ws;                                   // N floats
    float* colsum = ws + N;                               // 128 floats
    long   offA   = ((long)(N + 128) + 255) & ~255L;
    float* agg    = ws + offA;                            // N*D floats (also h2 output)
    float* h1     = agg + (long)N * D;                    // N*D floats
    (void)n_in; (void)out_size; (void)ws_size;

    // zero deg + colsum + agg
    long zcount = offA + (long)N * D;
    zero_f32<<<(int)((zcount + 255) / 256), 256, 0, stream>>>(ws, zcount);

    degree_kernel<<<(E + 255) / 256, 256, 0, stream>>>(dst, deg, E);

    long st = (long)E * 32;
    scatter_kernel<<<(int)((st + 255) / 256), 256, 0, stream>>>(x, src, dst, agg, E);

    // layer 1: h1 = relu(mean_agg @ W1l^T + b1 + x @ W1r^T)
    sage_gemm_kernel<<<N / 16, 256, 0, stream>>>(agg, deg, x, W1l, b1, W1r, h1);

    // layer 2 aggregation
    zero_f32<<<(int)(((long)N * D + 255) / 256), 256, 0, stream>>>(agg, (long)N * D);
    scatter_kernel<<<(int)((st + 255) / 256), 256, 0, stream>>>(h1, src, dst, agg, E);

    // layer 2: h2 written in place over agg (safe: rows are block-private, read pre-barrier)
    sage_gemm_kernel<<<N / 16, 256, 0, stream>>>(agg, deg, h1, W2l, b2, W2r, agg);

    float* outv = (float*)d_out;
    actor_kernel<<<(N + 7) / 8, 256, 0, stream>>>(agg, Wa, ba, outv, N);
    critic_partial_kernel<<<(N + 255) / 256, 128, 0, stream>>>(agg, colsum, N);
    critic_final_kernel<<<1, 128, 0, stream>>>(colsum, Wc, bc, outv + N, 1.0f / (float)N);
}